// LinearEncoder_24584392802336
// MI455X (gfx1250) — compile-verified
//
#include <hip/hip_runtime.h>
#include <hip/hip_bf16.h>

typedef __attribute__((ext_vector_type(2))) float v2f;
typedef __attribute__((ext_vector_type(8))) float v8f;

#define IN_CH 128
#define OUT_CH 128

// ---------------------------------------------------------------------------
// Degree kernels: deg[i] = 1 (self loop) + in-degree(i); then deg -> deg^-1/2
// ---------------------------------------------------------------------------
__global__ void deg_init_kernel(float* __restrict__ deg, int n) {
    int i = blockIdx.x * blockDim.x + threadIdx.x;
    if (i < n) deg[i] = 1.0f;  // self loop contributes 1 at every node
}

__global__ void deg_count_kernel(const int* __restrict__ dst, float* __restrict__ deg, int n_edges) {
    int i = blockIdx.x * blockDim.x + threadIdx.x;
    if (i < n_edges) atomicAdd(&deg[dst[i]], 1.0f);
}

__global__ void deg_rsqrt_kernel(float* __restrict__ deg, int n) {
    int i = blockIdx.x * blockDim.x + threadIdx.x;
    if (i < n) deg[i] = rsqrtf(deg[i]);  // deg >= 1 always, no zero guard needed
}

// ---------------------------------------------------------------------------
// h = x @ W via V_WMMA_F32_16X16X4_F32 (fp32 matrix pipe, wave32).
// One wave computes one 16x16 tile of h; 8 waves/block cover all 128 cols of
// a 16-row stripe. K = 128 unrolled in steps of 4 -> 32 WMMAs per wave.
// A 16x4 fragment: lane l holds row (l&15); VGPR0 = K + ((l>>4)?2:0), VGPR1 = +1.
// B 4x16 fragment mirrors A with N across lanes.  C/D: M = v + 8*(l>>4), N = l&15.
// ---------------------------------------------------------------------------
__global__ __launch_bounds__(256) void gcn_gemm_wmma_kernel(const float* __restrict__ x,
                                                            const float* __restrict__ W,
                                                            float* __restrict__ h,
                                                            int mtiles) {
    const int lane  = threadIdx.x & 31;
    const int ntile = threadIdx.x >> 5;       // 0..7  (8 * 16 = 128 output cols)
    const int mtile = blockIdx.x;
    if (mtile >= mtiles) return;

    const int mrow  = (lane & 15);
    const int col   = ntile * 16 + (lane & 15);
    const int khalf = (lane >> 4) << 1;       // 0 or 2

    const float* __restrict__ xrow = x + (size_t)(mtile * 16 + mrow) * IN_CH;

    v8f c = {};
#pragma unroll
    for (int k0 = 0; k0 < IN_CH; k0 += 4) {
        v2f a, bm;
        a.x  = xrow[k0 + khalf + 0];
        a.y  = xrow[k0 + khalf + 1];
        bm.x = W[(size_t)(k0 + khalf + 0) * OUT_CH + col];
        bm.y = W[(size_t)(k0 + khalf + 1) * OUT_CH + col];
        // 8 args: (neg_a, A, neg_b, B, c_mod, C, reuse_a, reuse_b)
        c = __builtin_amdgcn_wmma_f32_16x16x4_f32(false, a, false, bm,
                                                  (short)0, c, false, false);
    }

    float* __restrict__ hout = h + (size_t)(mtile * 16) * OUT_CH + ntile * 16 + (lane & 15);
    const int rbase = (lane >> 4) << 3;       // 0 or 8
#pragma unroll
    for (int v = 0; v < 8; ++v)
        hout[(size_t)(rbase + v) * OUT_CH] = c[v];
}

// Scalar tail for M % 16 != 0 (not hit for N_NODES = 100000, kept for safety).
__global__ void gcn_gemm_tail_kernel(const float* __restrict__ x, const float* __restrict__ W,
                                     float* __restrict__ h, int row_begin, int row_end) {
    int t   = blockIdx.x * blockDim.x + threadIdx.x;
    int row = row_begin + t / OUT_CH;
    int col = t % OUT_CH;
    if (row >= row_end) return;
    float acc = 0.0f;
    for (int k = 0; k < IN_CH; ++k) acc += x[(size_t)row * IN_CH + k] * W[(size_t)k * OUT_CH + col];
    h[(size_t)row * OUT_CH + col] = acc;
}

// ---------------------------------------------------------------------------
// out[i] = dinv[i]^2 * h[i] + b   (self-loop message; also initializes d_out).
// One wave per node: lane l handles channels 4l..4l+3 (coalesced float4).
// ---------------------------------------------------------------------------
__global__ __launch_bounds__(256) void gcn_self_init_kernel(const float* __restrict__ h,
                                                            const float* __restrict__ dinv,
                                                            const float* __restrict__ b,
                                                            float* __restrict__ out, int n_nodes) {
    const int lane = threadIdx.x & 31;
    const int node = blockIdx.x * (blockDim.x >> 5) + (threadIdx.x >> 5);
    if (node >= n_nodes) return;
    const float di = dinv[node];
    const float scale = di * di;
    const float4 hv = *(const float4*)(h + (size_t)node * OUT_CH + lane * 4);
    const float4 bv = *(const float4*)(b + lane * 4);
    float4 r;
    r.x = hv.x * scale + bv.x;
    r.y = hv.y * scale + bv.y;
    r.z = hv.z * scale + bv.z;
    r.w = hv.w * scale + bv.w;
    *(float4*)(out + (size_t)node * OUT_CH + lane * 4) = r;
}

// ---------------------------------------------------------------------------
// Edge scatter: out[dst] += dinv[src]*dinv[dst] * h[src].
// One wave per edge iteration; lane reads float4 of h[src] (512B coalesced row,
// L2-resident), 4 global_atomic_add_f32 into out[dst]. Grid-stride loop with
// global_prefetch_b8 of the next source row to hide L2 latency.
// ---------------------------------------------------------------------------
__global__ __launch_bounds__(256) void gcn_edge_scatter_kernel(const int* __restrict__ src,
                                                               const int* __restrict__ dst,
                                                               const float* __restrict__ h,
                                                               const float* __restrict__ dinv,
                                                               float* __restrict__ out,
                                                               int n_edges) {
    const int lane = threadIdx.x & 31;
    const long long wid    = (long long)blockIdx.x * (blockDim.x >> 5) + (threadIdx.x >> 5);
    const long long nwaves = (long long)gridDim.x * (blockDim.x >> 5);

    for (long long e = wid; e < n_edges; e += nwaves) {
        const int s = src[e];
        const int d = dst[e];

        const long long en = e + nwaves;     // prefetch next iteration's gather row
        if (en < n_edges) {
            const int sn = src[en];
            __builtin_prefetch(h + (size_t)sn * OUT_CH + lane * 4, 0, 1);
        }

        const float norm = dinv[s] * dinv[d];
        const float4 hv = *(const float4*)(h + (size_t)s * OUT_CH + lane * 4);
        float* __restrict__ orow = out + (size_t)d * OUT_CH + lane * 4;
        atomicAdd(orow + 0, norm * hv.x);
        atomicAdd(orow + 1, norm * hv.y);
        atomicAdd(orow + 2, norm * hv.z);
        atomicAdd(orow + 3, norm * hv.w);
    }
}

// ---------------------------------------------------------------------------
extern "C" void kernel_launch(void* const* d_in, const int* in_sizes, int n_in,
                              void* d_out, int out_size, void* d_ws, size_t ws_size,
                              hipStream_t stream) {
    const float* x  = (const float*)d_in[0];          // [N, 128]
    const int*   ei = (const int*)d_in[1];            // [2, E] (int32)
    const float* W  = (const float*)d_in[2];          // [128, 128]
    const float* b  = (const float*)d_in[3];          // [128]
    float* out      = (float*)d_out;                  // [N, 128]

    const int N = in_sizes[0] / IN_CH;
    const int E = in_sizes[1] / 2;
    const int* src = ei;
    const int* dst = ei + E;

    // Workspace layout: h [N*128 f32] then deg/dinv [N f32]
    float* h   = (float*)d_ws;
    float* deg = h + (size_t)N * OUT_CH;

    // 1) degrees -> deg^-1/2
    deg_init_kernel<<<(N + 255) / 256, 256, 0, stream>>>(deg, N);
    deg_count_kernel<<<(E + 255) / 256, 256, 0, stream>>>(dst, deg, E);
    deg_rsqrt_kernel<<<(N + 255) / 256, 256, 0, stream>>>(deg, N);

    // 2) h = x @ W  (fp32 WMMA)
    const int mtiles = N / 16;
    if (mtiles > 0)
        gcn_gemm_wmma_kernel<<<mtiles, 256, 0, stream>>>(x, W, h, mtiles);
    const int rem = N - mtiles * 16;
    if (rem > 0)
        gcn_gemm_tail_kernel<<<(rem * OUT_CH + 255) / 256, 256, 0, stream>>>(x, W, h, mtiles * 16, N);

    // 3) initialize out with self-loop message + bias
    gcn_self_init_kernel<<<(N + 7) / 8, 256, 0, stream>>>(h, deg, b, out, N);

    // 4) scatter edge messages (atomic f32 adds, L2-resident)
    gcn_edge_scatter_kernel<<<8192, 256, 0, stream>>>(src, dst, h, deg, out, E);
}